// observation_processing_network_90237262889560
// MI455X (gfx1250) — compile-verified
//
#include <hip/hip_runtime.h>
#include <hip/hip_bf16.h>

typedef __attribute__((ext_vector_type(2))) float v2f;
typedef __attribute__((ext_vector_type(8))) float v8f;

#define NEG_SLOPE 0.2f

__device__ __forceinline__ unsigned fenc(float f) {
    unsigned u = __float_as_uint(f);
    return (u & 0x80000000u) ? ~u : (u | 0x80000000u);
}
__device__ __forceinline__ float fdec(unsigned e) {
    return __uint_as_float((e & 0x80000000u) ? (e & 0x7FFFFFFFu) : ~e);
}

// ---------------- GAT ----------------
// Per-wave: 16 nodes, xp = h @ W via V_WMMA_F32_16X16X4_F32 (K=4, col 3 padded 0).
__global__ void gat_node_kernel(const float* __restrict__ hin, const float* __restrict__ W,
                                const float* __restrict__ asrc, const float* __restrict__ adst,
                                float* __restrict__ xp, float* __restrict__ sV, float* __restrict__ tV,
                                unsigned* __restrict__ maxenc, float* __restrict__ z,
                                float* __restrict__ acc, int N) {
    __shared__ float lxp[128 * 3];
    const int tid = threadIdx.x;
    const int lane = tid & 31;
    const int wave = tid >> 5;
    const int base = blockIdx.x * 128 + wave * 16;
    const int j15 = lane & 15;
    const bool full = (base + 16 <= N);   // uniform per wave: whole 16-node tile in range

    // A: 16x4 tile of h (rows = nodes, K = features, K=3 padded zero)
    const int row = base + j15;
    v2f a; a.x = 0.f; a.y = 0.f;
    if (full) {
        if (lane < 16) { a.x = hin[row * 3 + 0]; a.y = hin[row * 3 + 1]; }
        else           { a.x = hin[row * 3 + 2]; }
    } else if (row < N) {
        if (lane < 16) { a.x = hin[row * 3 + 0]; a.y = hin[row * 3 + 1]; }
        else           { a.x = hin[row * 3 + 2]; }
    }
    // B: 4x16 tile of W (3x3 zero padded)
    v2f b; b.x = 0.f; b.y = 0.f;
    if (lane < 16) {
        if (j15 < 3) { b.x = W[0 * 3 + j15]; b.y = W[1 * 3 + j15]; }
    } else {
        if (j15 < 3) { b.x = W[2 * 3 + j15]; }
    }
    v8f c = {};
    c = __builtin_amdgcn_wmma_f32_16x16x4_f32(false, a, false, b, (short)0, c, false, false);

    // D layout: VGPR j -> rows M=j (lanes 0-15, N=lane) and M=8+j (lanes 16-31, N=lane-16)
    if (j15 < 3) {
        const int r0 = base + ((lane < 16) ? 0 : 8);
        const int lr0 = wave * 16 + ((lane < 16) ? 0 : 8);
        if (full) {   // straight-line stores, no per-element exec masking
#pragma unroll
            for (int j = 0; j < 8; ++j) {
                float v = c[j];
                xp[(r0 + j) * 3 + j15] = v;
                lxp[(lr0 + j) * 3 + j15] = v;
            }
        } else {
#pragma unroll
            for (int j = 0; j < 8; ++j) {
                float v = c[j];
                if (r0 + j < N) xp[(r0 + j) * 3 + j15] = v;
                lxp[(lr0 + j) * 3 + j15] = v;
            }
        }
    }
    __syncthreads();
    if (tid < 128) {
        const int n = blockIdx.x * 128 + tid;
        if (n < N) {
            float x0 = lxp[tid * 3 + 0], x1 = lxp[tid * 3 + 1], x2 = lxp[tid * 3 + 2];
            sV[n] = x0 * asrc[0] + x1 * asrc[1] + x2 * asrc[2];
            tV[n] = x0 * adst[0] + x1 * adst[1] + x2 * adst[2];
            maxenc[n] = 0u;          // encoded -inf
            z[n] = 0.f;
            acc[n * 3 + 0] = 0.f; acc[n * 3 + 1] = 0.f; acc[n * 3 + 2] = 0.f;
        }
    }
}

__global__ void gat_edge_max_kernel(const int* __restrict__ ei, const float* __restrict__ sV,
                                    const float* __restrict__ tV, unsigned* __restrict__ maxenc,
                                    int E, int N) {
    const int i = blockIdx.x * blockDim.x + threadIdx.x;
    if (i >= E + N) return;
    const int src = (i < E) ? ei[i] : (i - E);
    const int dst = (i < E) ? ei[E + i] : (i - E);
    float e = sV[src] + tV[dst];
    e = (e >= 0.f) ? e : NEG_SLOPE * e;
    atomicMax(&maxenc[dst], fenc(e));
}

__global__ void gat_edge_acc_kernel(const int* __restrict__ ei, const float* __restrict__ sV,
                                    const float* __restrict__ tV, const float* __restrict__ xp,
                                    const unsigned* __restrict__ maxenc, float* __restrict__ z,
                                    float* __restrict__ acc, int E, int N) {
    const int i = blockIdx.x * blockDim.x + threadIdx.x;
    if (i >= E + N) return;
    const int src = (i < E) ? ei[i] : (i - E);
    const int dst = (i < E) ? ei[E + i] : (i - E);
    float e = sV[src] + tV[dst];
    e = (e >= 0.f) ? e : NEG_SLOPE * e;
    const float w = __expf(e - fdec(maxenc[dst]));
    atomicAdd(&z[dst], w);
    atomicAdd(&acc[dst * 3 + 0], w * xp[src * 3 + 0]);
    atomicAdd(&acc[dst * 3 + 1], w * xp[src * 3 + 1]);
    atomicAdd(&acc[dst * 3 + 2], w * xp[src * 3 + 2]);
}

__global__ void gat_finalize_kernel(const float* __restrict__ acc, const float* __restrict__ z,
                                    const float* __restrict__ bias, float* __restrict__ h,
                                    int N, int do_relu) {
    const int n = blockIdx.x * blockDim.x + threadIdx.x;
    if (n >= N) return;
    const float inv = 1.f / (z[n] + 1e-16f);
#pragma unroll
    for (int cidx = 0; cidx < 3; ++cidx) {
        float v = acc[n * 3 + cidx] * inv + bias[cidx];
        if (do_relu) v = fmaxf(v, 0.f);
        h[n * 3 + cidx] = v;
    }
}

// ---------------- MHA (head dim 1 => row-max via per-head global k min/max) ---------
__global__ void mha_init_kernel(unsigned* kmm) {
    const int t = threadIdx.x;
    if (t < 3) kmm[t] = 0u;                 // max slots: encoded minimum
    else if (t < 6) kmm[t] = 0xFFFFFFFFu;   // min slots: encoded maximum
}

__global__ void mha_qkv_kernel(const float* __restrict__ h, const float* __restrict__ in_w,
                               const float* __restrict__ in_b, float* __restrict__ qkv,
                               unsigned* __restrict__ kmm, int N) {
    const int n = blockIdx.x * blockDim.x + threadIdx.x;
    const bool ok = (n < N);
    float h0 = 0.f, h1 = 0.f, h2v = 0.f;
    if (ok) { h0 = h[n * 3 + 0]; h1 = h[n * 3 + 1]; h2v = h[n * 3 + 2]; }
    float kl[3];
#pragma unroll
    for (int j = 0; j < 9; ++j) {
        float v = h0 * in_w[j * 3 + 0] + h1 * in_w[j * 3 + 1] + h2v * in_w[j * 3 + 2] + in_b[j];
        if (ok) qkv[n * 9 + j] = v;
        if (j >= 3 && j < 6) kl[j - 3] = v;
    }
    // per-wave min/max reduce, then one atomic per wave per slot
#pragma unroll
    for (int hh = 0; hh < 3; ++hh) {
        float vmax = ok ? kl[hh] : -3.0e38f;
        float vmin = ok ? kl[hh] : 3.0e38f;
        for (int o = 16; o > 0; o >>= 1) {
            vmax = fmaxf(vmax, __shfl_xor(vmax, o));
            vmin = fminf(vmin, __shfl_xor(vmin, o));
        }
        if ((threadIdx.x & 31) == 0) {
            atomicMax(&kmm[hh], fenc(vmax));
            atomicMin(&kmm[3 + hh], fenc(vmin));
        }
    }
}

__global__ void mha_attn_kernel(const float* __restrict__ qkv, const unsigned* __restrict__ kmm,
                                const float* __restrict__ out_w, const float* __restrict__ out_b,
                                float* __restrict__ h2, int N) {
    const int n = blockIdx.x;
    const int tid = threadIdx.x;
    float aq[3], Kc[3];
#pragma unroll
    for (int hh = 0; hh < 3; ++hh) {
        aq[hh] = qkv[n * 9 + hh];
        Kc[hh] = (aq[hh] >= 0.f) ? fdec(kmm[hh]) : fdec(kmm[3 + hh]);
    }
    float num[3] = {0.f, 0.f, 0.f}, den[3] = {0.f, 0.f, 0.f};
    for (int m = tid; m < N; m += blockDim.x) {
        const float* q = &qkv[m * 9];
#pragma unroll
        for (int hh = 0; hh < 3; ++hh) {
            float w = __expf(aq[hh] * (q[3 + hh] - Kc[hh]));
            num[hh] += w * q[6 + hh];
            den[hh] += w;
        }
    }
    __shared__ float sred[6];
    if (tid < 6) sred[tid] = 0.f;
    __syncthreads();
    float vals[6] = {num[0], num[1], num[2], den[0], den[1], den[2]};
#pragma unroll
    for (int k = 0; k < 6; ++k) {
        float v = vals[k];
        for (int o = 16; o > 0; o >>= 1) v += __shfl_xor(v, o);
        if ((tid & 31) == 0) atomicAdd(&sred[k], v);
    }
    __syncthreads();
    if (tid == 0) {
        float o0 = sred[0] / sred[3], o1 = sred[1] / sred[4], o2 = sred[2] / sred[5];
#pragma unroll
        for (int cc = 0; cc < 3; ++cc)
            h2[n * 3 + cc] = o0 * out_w[cc * 3 + 0] + o1 * out_w[cc * 3 + 1] +
                             o2 * out_w[cc * 3 + 2] + out_b[cc];
    }
}

// ---------------- Transformer-conv ----------------
__global__ void tc_node_kernel(const float* __restrict__ h2,
                               const float* Wq, const float* bq, const float* Wk, const float* bk,
                               const float* Wv, const float* bv, const float* Ws, const float* bs,
                               float* __restrict__ q2, float* __restrict__ k2,
                               float* __restrict__ v2, float* __restrict__ skp,
                               unsigned* __restrict__ maxenc, float* __restrict__ z,
                               float* __restrict__ acc, int N) {
    const int n = blockIdx.x * blockDim.x + threadIdx.x;
    if (n >= N) return;
    const float h0 = h2[n * 3 + 0], h1 = h2[n * 3 + 1], hc2 = h2[n * 3 + 2];
#pragma unroll
    for (int j = 0; j < 3; ++j) {
        q2[n * 3 + j]  = h0 * Wq[0 * 3 + j] + h1 * Wq[1 * 3 + j] + hc2 * Wq[2 * 3 + j] + bq[j];
        k2[n * 3 + j]  = h0 * Wk[0 * 3 + j] + h1 * Wk[1 * 3 + j] + hc2 * Wk[2 * 3 + j] + bk[j];
        v2[n * 3 + j]  = h0 * Wv[0 * 3 + j] + h1 * Wv[1 * 3 + j] + hc2 * Wv[2 * 3 + j] + bv[j];
        skp[n * 3 + j] = h0 * Ws[0 * 3 + j] + h1 * Ws[1 * 3 + j] + hc2 * Ws[2 * 3 + j] + bs[j];
        acc[n * 3 + j] = 0.f;
    }
    maxenc[n] = 0u;
    z[n] = 0.f;
}

__global__ void tc_edge_max_kernel(const int* __restrict__ ei, const float* __restrict__ q2,
                                   const float* __restrict__ k2, unsigned* __restrict__ maxenc,
                                   int E, int N) {
    const int i = blockIdx.x * blockDim.x + threadIdx.x;
    if (i >= E + N) return;
    const int src = (i < E) ? ei[i] : (i - E);
    const int dst = (i < E) ? ei[E + i] : (i - E);
    const float lg = (q2[dst * 3 + 0] * k2[src * 3 + 0] + q2[dst * 3 + 1] * k2[src * 3 + 1] +
                      q2[dst * 3 + 2] * k2[src * 3 + 2]) * 0.57735026919f;
    atomicMax(&maxenc[dst], fenc(lg));
}

__global__ void tc_edge_acc_kernel(const int* __restrict__ ei, const float* __restrict__ q2,
                                   const float* __restrict__ k2, const float* __restrict__ v2,
                                   const unsigned* __restrict__ maxenc, float* __restrict__ z,
                                   float* __restrict__ acc, int E, int N) {
    const int i = blockIdx.x * blockDim.x + threadIdx.x;
    if (i >= E + N) return;
    const int src = (i < E) ? ei[i] : (i - E);
    const int dst = (i < E) ? ei[E + i] : (i - E);
    const float lg = (q2[dst * 3 + 0] * k2[src * 3 + 0] + q2[dst * 3 + 1] * k2[src * 3 + 1] +
                      q2[dst * 3 + 2] * k2[src * 3 + 2]) * 0.57735026919f;
    const float w = __expf(lg - fdec(maxenc[dst]));
    atomicAdd(&z[dst], w);
    atomicAdd(&acc[dst * 3 + 0], w * v2[src * 3 + 0]);
    atomicAdd(&acc[dst * 3 + 1], w * v2[src * 3 + 1]);
    atomicAdd(&acc[dst * 3 + 2], w * v2[src * 3 + 2]);
}

__global__ void tc_finalize_kernel(const float* __restrict__ acc, const float* __restrict__ z,
                                   const float* __restrict__ skp, float* __restrict__ h3,
                                   float* __restrict__ a1pre, int N) {
    const int n = blockIdx.x * blockDim.x + threadIdx.x;
    if (blockIdx.x == 0 && threadIdx.x < 16) a1pre[threadIdx.x] = 0.f;
    if (n >= N) return;
    const float inv = 1.f / (z[n] + 1e-16f);
#pragma unroll
    for (int cidx = 0; cidx < 3; ++cidx)
        h3[n * 3 + cidx] = acc[n * 3 + cidx] * inv + skp[n * 3 + cidx];
}

// ---------------- MLP layer 1: (1 x K) x (K x 16) GEMV via WMMA K-chunks ----------------
__device__ __forceinline__ void mlp1_load(const float* __restrict__ flat,
                                          const float* __restrict__ W1, int k0,
                                          int lane, int j15, v2f& a, v2f& b) {
    a.x = 0.f; a.y = 0.f;                     // A row 0 = flat, rows 1..15 = 0
    if (lane == 0)       { a.x = flat[k0 + 0]; a.y = flat[k0 + 1]; }
    else if (lane == 16) { a.x = flat[k0 + 2]; a.y = flat[k0 + 3]; }
    if (lane < 16) { b.x = W1[(k0 + 0) * 16 + j15]; b.y = W1[(k0 + 1) * 16 + j15]; }
    else           { b.x = W1[(k0 + 2) * 16 + j15]; b.y = W1[(k0 + 3) * 16 + j15]; }
}

__global__ void mlp1_wmma_kernel(const float* __restrict__ flat, const float* __restrict__ W1,
                                 float* __restrict__ a1pre, int K) {
    const int lane = threadIdx.x & 31;
    const int j15 = lane & 15;
    const int gwave = blockIdx.x * (blockDim.x >> 5) + (threadIdx.x >> 5);
    const int nwaves = gridDim.x * (blockDim.x >> 5);
    const int chunks = K >> 2;                 // K multiple of 4
    const int stride4 = nwaves * 4;
    const int blocked = (chunks / stride4) * stride4;

    v8f c0 = {}, c1 = {}, c2 = {}, c3 = {};    // 4 independent accumulators: no D->C stall chain
    v2f a, b;
    for (int cb = gwave * 4; cb < blocked; cb += stride4) {
        mlp1_load(flat, W1, (cb + 0) << 2, lane, j15, a, b);
        c0 = __builtin_amdgcn_wmma_f32_16x16x4_f32(false, a, false, b, (short)0, c0, false, false);
        mlp1_load(flat, W1, (cb + 1) << 2, lane, j15, a, b);
        c1 = __builtin_amdgcn_wmma_f32_16x16x4_f32(false, a, false, b, (short)0, c1, false, false);
        mlp1_load(flat, W1, (cb + 2) << 2, lane, j15, a, b);
        c2 = __builtin_amdgcn_wmma_f32_16x16x4_f32(false, a, false, b, (short)0, c2, false, false);
        mlp1_load(flat, W1, (cb + 3) << 2, lane, j15, a, b);
        c3 = __builtin_amdgcn_wmma_f32_16x16x4_f32(false, a, false, b, (short)0, c3, false, false);
    }
    for (int ch = blocked + gwave; ch < chunks; ch += nwaves) {   // generic remainder
        mlp1_load(flat, W1, ch << 2, lane, j15, a, b);
        c0 = __builtin_amdgcn_wmma_f32_16x16x4_f32(false, a, false, b, (short)0, c0, false, false);
    }
    if (lane < 16)   // D row M=0 lives in element 0, lanes 0-15
        atomicAdd(&a1pre[lane], c0[0] + c1[0] + c2[0] + c3[0]);
}

__global__ void mlp2_kernel(const float* __restrict__ a1pre, const float* __restrict__ b1,
                            const float* __restrict__ W2, const float* __restrict__ b2,
                            float* __restrict__ a2, float* __restrict__ ressum) {
    __shared__ float a1[16];
    const int t = threadIdx.x;
    if (t < 16) a1[t] = fmaxf(a1pre[t] + b1[t], 0.f);
    __syncthreads();
    if (t < 32) {
        float v = b2[t];
#pragma unroll
        for (int i = 0; i < 16; ++i) v += a1[i] * W2[i * 32 + t];
        a2[t] = fmaxf(v, 0.f);
    }
    if (t == 0) *ressum = 0.f;
}

__global__ void mlp3_kernel(const float* __restrict__ a2, const float* __restrict__ W3,
                            const float* __restrict__ b3, const float* __restrict__ mask,
                            float* __restrict__ out, float* __restrict__ ressum, int N) {
    const int j = blockIdx.x * blockDim.x + threadIdx.x;
    float r = 0.f;
    if (j < N) {
        __builtin_prefetch(&W3[j], 0, 0);  // global_prefetch_b8
        r = b3[j];
#pragma unroll
        for (int k = 0; k < 32; ++k) r += a2[k] * W3[k * N + j];
        out[j] = r * mask[j];
    }
    float v = r;
    for (int o = 16; o > 0; o >>= 1) v += __shfl_xor(v, o);
    if ((threadIdx.x & 31) == 0) atomicAdd(ressum, v);
}

__global__ void value_kernel(const float* __restrict__ ressum, const float* __restrict__ cW,
                             const float* __restrict__ cb, float* __restrict__ out, int N) {
    if (threadIdx.x == 0 && blockIdx.x == 0)
        out[N] = cW[0] * (ressum[0] / (float)N) + cb[0];
}

extern "C" void kernel_launch(void* const* d_in, const int* in_sizes, int n_in,
                              void* d_out, int out_size, void* d_ws, size_t ws_size,
                              hipStream_t stream) {
    const float* x        = (const float*)d_in[0];
    const float* mask     = (const float*)d_in[1];
    const int*   ei       = (const int*)d_in[2];
    const float* gat_W    = (const float*)d_in[3];
    const float* gat_asrc = (const float*)d_in[4];
    const float* gat_adst = (const float*)d_in[5];
    const float* gat_b    = (const float*)d_in[6];
    const float* mha_in_w = (const float*)d_in[7];
    const float* mha_in_b = (const float*)d_in[8];
    const float* mha_out_w= (const float*)d_in[9];
    const float* mha_out_b= (const float*)d_in[10];
    const float* tc_Wq    = (const float*)d_in[11];
    const float* tc_bq    = (const float*)d_in[12];
    const float* tc_Wk    = (const float*)d_in[13];
    const float* tc_bk    = (const float*)d_in[14];
    const float* tc_Wv    = (const float*)d_in[15];
    const float* tc_bv    = (const float*)d_in[16];
    const float* tc_Ws    = (const float*)d_in[17];
    const float* tc_bs    = (const float*)d_in[18];
    const float* mlp_W1   = (const float*)d_in[19];
    const float* mlp_b1   = (const float*)d_in[20];
    const float* mlp_W2   = (const float*)d_in[21];
    const float* mlp_b2   = (const float*)d_in[22];
    const float* mlp_W3   = (const float*)d_in[23];
    const float* mlp_b3   = (const float*)d_in[24];
    const float* critic_W = (const float*)d_in[25];
    const float* critic_b = (const float*)d_in[26];

    const int N = in_sizes[1];
    const int E = in_sizes[2] / 2;
    const int LG = 10;

    float* ws = (float*)d_ws;
    float* h       = ws; ws += (size_t)N * 3;
    float* xp      = ws; ws += (size_t)N * 3;
    float* sV      = ws; ws += N;
    float* tV      = ws; ws += N;
    unsigned* maxe = (unsigned*)ws; ws += N;
    float* z       = ws; ws += N;
    float* acc     = ws; ws += (size_t)N * 3;
    float* qkv     = ws; ws += (size_t)N * 9;
    float* h2      = ws; ws += (size_t)N * 3;
    float* q2      = ws; ws += (size_t)N * 3;
    float* k2      = ws; ws += (size_t)N * 3;
    float* v2      = ws; ws += (size_t)N * 3;
    float* skp     = ws; ws += (size_t)N * 3;
    float* h3      = ws; ws += (size_t)N * 3;
    unsigned* kmm  = (unsigned*)ws; ws += 8;
    float* a1pre   = ws; ws += 16;
    float* a2buf   = ws; ws += 32;
    float* ressum  = ws; ws += 1;

    float* out = (float*)d_out;

    const int T = 256;
    const int nodeBlocks = (N + 127) / 128;            // WMMA: 8 waves x 16 nodes per block
    const int nodeGrid   = (N + T - 1) / T;
    const int edgeGrid   = (E + N + T - 1) / T;

    for (int l = 0; l < LG; ++l) {
        const float* hin = (l == 0) ? x : h;
        gat_node_kernel<<<nodeBlocks, T, 0, stream>>>(hin, gat_W + 9 * l, gat_asrc + 3 * l,
                                                      gat_adst + 3 * l, xp, sV, tV, maxe, z, acc, N);
        gat_edge_max_kernel<<<edgeGrid, T, 0, stream>>>(ei, sV, tV, maxe, E, N);
        gat_edge_acc_kernel<<<edgeGrid, T, 0, stream>>>(ei, sV, tV, xp, maxe, z, acc, E, N);
        gat_finalize_kernel<<<nodeGrid, T, 0, stream>>>(acc, z, gat_b + 3 * l, h, N,
                                                        (l < LG - 1) ? 1 : 0);
    }
    mha_init_kernel<<<1, 32, 0, stream>>>(kmm);
    mha_qkv_kernel<<<nodeGrid, T, 0, stream>>>(h, mha_in_w, mha_in_b, qkv, kmm, N);
    mha_attn_kernel<<<N, T, 0, stream>>>(qkv, kmm, mha_out_w, mha_out_b, h2, N);
    tc_node_kernel<<<nodeGrid, T, 0, stream>>>(h2, tc_Wq, tc_bq, tc_Wk, tc_bk, tc_Wv, tc_bv,
                                               tc_Ws, tc_bs, q2, k2, v2, skp, maxe, z, acc, N);
    tc_edge_max_kernel<<<edgeGrid, T, 0, stream>>>(ei, q2, k2, maxe, E, N);
    tc_edge_acc_kernel<<<edgeGrid, T, 0, stream>>>(ei, q2, k2, v2, maxe, z, acc, E, N);
    tc_finalize_kernel<<<nodeGrid, T, 0, stream>>>(acc, z, skp, h3, a1pre, N);
    mlp1_wmma_kernel<<<24, T, 0, stream>>>(h3, mlp_W1, a1pre, 3 * N);
    mlp2_kernel<<<1, 64, 0, stream>>>(a1pre, mlp_b1, mlp_W2, mlp_b2, a2buf, ressum);
    mlp3_kernel<<<(N + T - 1) / T, T, 0, stream>>>(a2buf, mlp_W3, mlp_b3, mask, out, ressum, N);
    value_kernel<<<1, 1, 0, stream>>>(ressum, critic_W, critic_b, out, N);
}